// Decoder_26826365731370
// MI455X (gfx1250) — compile-verified
//
#include <hip/hip_runtime.h>
#include <math.h>

#define V_SIZE 32000
#define H_SIZE 1024
#define E_SIZE 512
#define B_SIZE 64
#define S_SIZE 128
#define XK     (E_SIZE + H_SIZE)   // 1536
#define G4     (4 * H_SIZE)        // 4096

typedef __attribute__((ext_vector_type(2))) float v2f;
typedef __attribute__((ext_vector_type(8))) float v8f;

__device__ __forceinline__ v8f wmma4(v2f a, v2f b, v8f c) {
  // D(16x16,f32) = A(16x4,f32) x B(4x16,f32) + C   -> v_wmma_f32_16x16x4_f32
  return __builtin_amdgcn_wmma_f32_16x16x4_f32(false, a, false, b, (short)0, c,
                                               false, false);
}

// ---------------------------------------------------------------- embeddings
__global__ void k_gather_emb(const int* __restrict__ idx,
                             const float* __restrict__ emb,
                             float* __restrict__ x) {
  int t = blockIdx.x * blockDim.x + threadIdx.x;
  if (t >= B_SIZE * E_SIZE) return;
  int b = t / E_SIZE, e = t - b * E_SIZE;
  x[b * XK + e] = emb[(size_t)idx[b] * E_SIZE + e];
}

// --------------------------------------------------- attention scores+softmax
__global__ void k_attn_softmax(const float* __restrict__ enc,     // (B,S,H)
                               const float* __restrict__ prev_h,  // (B,H)
                               const float* __restrict__ attn_w,  // (1,2H)
                               const float* __restrict__ attn_b,  // (1,)
                               const int* __restrict__ curr_idxs, // (B,)
                               float* __restrict__ attn_out) {    // (B,S)
  const int b    = blockIdx.x;
  const int tid  = threadIdx.x;      // 128 threads = 4 waves
  const int lane = tid & 31;
  const int wave = tid >> 5;
  __shared__ float sh_scores[S_SIZE];
  __shared__ float sh_red[4];
  __shared__ float sh_tmp[S_SIZE];
  __shared__ int   sh_mask;

  // hb = dot(prev_h[b], w_h)
  float hb = 0.f;
  for (int h = tid; h < H_SIZE; h += 128)
    hb += prev_h[b * H_SIZE + h] * attn_w[h];
  for (int off = 16; off > 0; off >>= 1) hb += __shfl_down(hb, off, 32);
  if (lane == 0) sh_red[wave] = hb;
  if (tid == 0) sh_mask = 0;
  __syncthreads();
  const float hbsum = sh_red[0] + sh_red[1] + sh_red[2] + sh_red[3];
  if (tid < B_SIZE && curr_idxs[tid] == b) atomicOr(&sh_mask, 1);

  // per-s dot products: each wave covers 32 s values, lanes split H
  const float* encb = enc + (size_t)b * S_SIZE * H_SIZE;
  const float* w_e  = attn_w + H_SIZE;
  for (int s = wave * 32; s < wave * 32 + 32; ++s) {
    float acc = 0.f;
    const float* row = encb + (size_t)s * H_SIZE;
    for (int h = lane; h < H_SIZE; h += 32) acc += row[h] * w_e[h];
    for (int off = 16; off > 0; off >>= 1) acc += __shfl_down(acc, off, 32);
    if (lane == 0) sh_scores[s] = acc + hbsum + attn_b[0];
  }
  __syncthreads();

  const bool masked_out = (sh_mask == 0);
  float val = masked_out ? 0.f : sh_scores[tid];   // where(mask, scores, 0)

  // softmax over S (blockDim == S == 128)
  sh_tmp[tid] = val; __syncthreads();
  for (int off = 64; off > 0; off >>= 1) {
    if (tid < off) sh_tmp[tid] = fmaxf(sh_tmp[tid], sh_tmp[tid + off]);
    __syncthreads();
  }
  const float mx = sh_tmp[0]; __syncthreads();
  const float ex = expf(val - mx);
  sh_tmp[tid] = ex; __syncthreads();
  for (int off = 64; off > 0; off >>= 1) {
    if (tid < off) sh_tmp[tid] += sh_tmp[tid + off];
    __syncthreads();
  }
  attn_out[b * S_SIZE + tid] = ex / sh_tmp[0];
}

// ------------------------------------------------------------------- context
__global__ void k_context(const float* __restrict__ enc,
                          const float* __restrict__ attn,
                          float* __restrict__ x) {
  const int b   = blockIdx.x;
  const int tid = threadIdx.x;   // 256
  __shared__ float sh_a[S_SIZE];
  if (tid < S_SIZE) sh_a[tid] = attn[b * S_SIZE + tid];
  __syncthreads();
  const float* encb = enc + (size_t)b * S_SIZE * H_SIZE;
  float acc[4] = {0.f, 0.f, 0.f, 0.f};
  for (int s = 0; s < S_SIZE; ++s) {
    const float a = sh_a[s];
    const float* row = encb + (size_t)s * H_SIZE;
#pragma unroll
    for (int j = 0; j < 4; ++j) acc[j] += a * row[tid + j * 256];
  }
#pragma unroll
  for (int j = 0; j < 4; ++j) x[b * XK + E_SIZE + tid + j * 256] = acc[j];
}

// -------------------------------------------------------- gates GEMM (WMMA)
// gates(64,4096) = x(64,1536)@w_ih^T + prev_h(64,1024)@w_hh^T + b_ih + b_hh
__global__ __launch_bounds__(256) void k_gates_wmma(
    const float* __restrict__ x, const float* __restrict__ w_ih,
    const float* __restrict__ b_ih, const float* __restrict__ prev_h,
    const float* __restrict__ w_hh, const float* __restrict__ b_hh,
    float* __restrict__ gates) {
  const int lane = threadIdx.x & 31;
  const int wave = threadIdx.x >> 5;
  const int l16  = lane & 15;
  const int half = lane >> 4;
  const int n0   = blockIdx.x * 128 + wave * 16;  // wave owns 16 output cols

  const v8f zero = {0.f, 0.f, 0.f, 0.f, 0.f, 0.f, 0.f, 0.f};
  v8f acc[4] = {zero, zero, zero, zero};          // all 64 batch rows (4 M-tiles)

  for (int k = 0; k < XK; k += 4) {
    v2f bfrag = *(const v2f*)(w_ih + (size_t)(n0 + l16) * XK + k + 2 * half);
#pragma unroll
    for (int mt = 0; mt < 4; ++mt) {
      v2f afrag = *(const v2f*)(x + (size_t)(mt * 16 + l16) * XK + k + 2 * half);
      acc[mt] = wmma4(afrag, bfrag, acc[mt]);
    }
  }
  for (int k = 0; k < H_SIZE; k += 4) {
    v2f bfrag = *(const v2f*)(w_hh + (size_t)(n0 + l16) * H_SIZE + k + 2 * half);
#pragma unroll
    for (int mt = 0; mt < 4; ++mt) {
      v2f afrag = *(const v2f*)(prev_h + (size_t)(mt * 16 + l16) * H_SIZE + k + 2 * half);
      acc[mt] = wmma4(afrag, bfrag, acc[mt]);
    }
  }
  const int n = n0 + l16;
  const float bias = b_ih[n] + b_hh[n];
#pragma unroll
  for (int mt = 0; mt < 4; ++mt)
#pragma unroll
    for (int r = 0; r < 8; ++r) {
      const int m = mt * 16 + r + 8 * half;   // D layout: M = r + 8*half, N = l16
      gates[(size_t)m * G4 + n] = acc[mt][r] + bias;
    }
}

// ------------------------------------------------------------- LSTM pointwise
__global__ void k_lstm(const float* __restrict__ gates,
                       const float* __restrict__ prev_c,
                       float* __restrict__ hidden, float* __restrict__ cell) {
  int t = blockIdx.x * blockDim.x + threadIdx.x;
  if (t >= B_SIZE * H_SIZE) return;
  int b = t / H_SIZE, h = t - b * H_SIZE;
  const float* g = gates + (size_t)b * G4;
  const float gi = g[h], gf = g[H_SIZE + h], gg = g[2 * H_SIZE + h], go = g[3 * H_SIZE + h];
  const float si = 1.f / (1.f + expf(-gi));
  const float sf = 1.f / (1.f + expf(-gf));
  const float so = 1.f / (1.f + expf(-go));
  const float c  = sf * prev_c[t] + si * tanhf(gg);
  cell[t]   = c;
  hidden[t] = so * tanhf(c);
}

// ------------------------------------------------------- logits GEMM (WMMA)
// logits(64,32000) = hidden(64,1024)@out_w^T + out_b
__global__ __launch_bounds__(256) void k_logits_wmma(
    const float* __restrict__ hidden, const float* __restrict__ out_w,
    const float* __restrict__ out_b, float* __restrict__ logits) {
  const int lane = threadIdx.x & 31;
  const int wave = threadIdx.x >> 5;
  const int l16  = lane & 15;
  const int half = lane >> 4;
  const int n0   = blockIdx.x * 128 + wave * 16;

  const v8f zero = {0.f, 0.f, 0.f, 0.f, 0.f, 0.f, 0.f, 0.f};
  v8f acc[4] = {zero, zero, zero, zero};

  for (int k = 0; k < H_SIZE; k += 4) {
    v2f bfrag = *(const v2f*)(out_w + (size_t)(n0 + l16) * H_SIZE + k + 2 * half);
#pragma unroll
    for (int mt = 0; mt < 4; ++mt) {
      v2f afrag = *(const v2f*)(hidden + (size_t)(mt * 16 + l16) * H_SIZE + k + 2 * half);
      acc[mt] = wmma4(afrag, bfrag, acc[mt]);
    }
  }
  const int n = n0 + l16;
  const float bias = out_b[n];
#pragma unroll
  for (int mt = 0; mt < 4; ++mt)
#pragma unroll
    for (int r = 0; r < 8; ++r) {
      const int m = mt * 16 + r + 8 * half;
      logits[(size_t)m * V_SIZE + n] = acc[mt][r] + bias;
    }
}

// ------------------------------------------------------- in-place log_softmax
__global__ void k_logsoftmax(float* __restrict__ out) {
  const int b   = blockIdx.x;
  const int tid = threadIdx.x;  // 256
  float* row = out + (size_t)b * V_SIZE;
  __shared__ float sh[256];
  float m = -__builtin_inff();
  for (int v = tid; v < V_SIZE; v += 256) m = fmaxf(m, row[v]);
  sh[tid] = m; __syncthreads();
  for (int off = 128; off > 0; off >>= 1) {
    if (tid < off) sh[tid] = fmaxf(sh[tid], sh[tid + off]);
    __syncthreads();
  }
  m = sh[0]; __syncthreads();
  float s = 0.f;
  for (int v = tid; v < V_SIZE; v += 256) s += expf(row[v] - m);
  sh[tid] = s; __syncthreads();
  for (int off = 128; off > 0; off >>= 1) {
    if (tid < off) sh[tid] += sh[tid + off];
    __syncthreads();
  }
  const float lse = m + logf(sh[0]);
  for (int v = tid; v < V_SIZE; v += 256) row[v] -= lse;
}

// ---------------------------------------------------------------------------
extern "C" void kernel_launch(void* const* d_in, const int* in_sizes, int n_in,
                              void* d_out, int out_size, void* d_ws, size_t ws_size,
                              hipStream_t stream) {
  const int*   input_batch = (const int*)d_in[0];
  const float* prev_h      = (const float*)d_in[1];
  const float* prev_c      = (const float*)d_in[2];
  const float* enc         = (const float*)d_in[3];
  const int*   curr_idxs   = (const int*)d_in[4];
  const float* emb         = (const float*)d_in[5];
  const float* attn_w      = (const float*)d_in[6];
  const float* attn_b      = (const float*)d_in[7];
  const float* w_ih        = (const float*)d_in[8];
  const float* b_ih        = (const float*)d_in[9];
  const float* w_hh        = (const float*)d_in[10];
  const float* b_hh        = (const float*)d_in[11];
  const float* out_w       = (const float*)d_in[12];
  const float* out_b       = (const float*)d_in[13];

  // outputs concatenated flat: output(B,V), hidden(B,H), cell(B,H), attn(B,S)
  float* out        = (float*)d_out;
  float* out_logits = out;
  float* out_hidden = out + (size_t)B_SIZE * V_SIZE;
  float* out_cell   = out_hidden + B_SIZE * H_SIZE;
  float* out_attn   = out_cell + B_SIZE * H_SIZE;

  float* ws    = (float*)d_ws;
  float* x     = ws;                   // (B, XK)  = 98304 floats
  float* gates = x + B_SIZE * XK;      // (B, 4H)  = 262144 floats

  k_gather_emb<<<(B_SIZE * E_SIZE + 255) / 256, 256, 0, stream>>>(input_batch, emb, x);
  k_attn_softmax<<<B_SIZE, 128, 0, stream>>>(enc, prev_h, attn_w, attn_b, curr_idxs, out_attn);
  k_context<<<B_SIZE, 256, 0, stream>>>(enc, out_attn, x);
  k_gates_wmma<<<G4 / 128, 256, 0, stream>>>(x, w_ih, b_ih, prev_h, w_hh, b_hh, gates);
  k_lstm<<<(B_SIZE * H_SIZE + 255) / 256, 256, 0, stream>>>(gates, prev_c, out_hidden, out_cell);
  k_logits_wmma<<<V_SIZE / 128, 256, 0, stream>>>(out_hidden, out_w, out_b, out_logits);
  k_logsoftmax<<<B_SIZE, 256, 0, stream>>>(out_logits);
}